// DecoderIterative_22282290331755
// MI455X (gfx1250) — compile-verified
//
#include <hip/hip_runtime.h>

typedef __attribute__((ext_vector_type(2))) float v2f;
typedef __attribute__((ext_vector_type(8))) float v8f;
typedef __attribute__((ext_vector_type(4))) unsigned v4u;
typedef __attribute__((ext_vector_type(8))) unsigned v8u;

// ---------------------------------------------------------------------------
// Preamble kernels: build M = (I - L)^-1 by log-depth doubling (L nilpotent),
// then fold W2 = M * Wx. All tiny (~0.54 GFLOP total).
// ---------------------------------------------------------------------------

// M0 = I + L, T0 = L   (L = strict lower triangle of Wy)
__global__ __launch_bounds__(256) void init_MT(const float* __restrict__ Wy,
                                               float* __restrict__ M0,
                                               float* __restrict__ T0, int D) {
  int j = blockIdx.x * 16 + (threadIdx.x & 15);
  int i = blockIdx.y * 16 + (threadIdx.x >> 4);
  float l = (j < i) ? Wy[(size_t)i * D + j] : 0.0f;
  M0[(size_t)i * D + j] = l + ((i == j) ? 1.0f : 0.0f);
  T0[(size_t)i * D + j] = l;
}

// C = A(Md x Kd) * B(Kd x Nd)  [+ A if addA, valid when Nd <= Kd]
__global__ __launch_bounds__(256) void matmul_nn(const float* __restrict__ A,
                                                 const float* __restrict__ B,
                                                 float* __restrict__ C,
                                                 int Md, int Nd, int Kd, int addA) {
  int j = blockIdx.x * 16 + (threadIdx.x & 15);
  int i = blockIdx.y * 16 + (threadIdx.x >> 4);
  if (i >= Md || j >= Nd) return;
  float s = addA ? A[(size_t)i * Kd + j] : 0.0f;
  for (int k = 0; k < Kd; ++k)
    s += A[(size_t)i * Kd + k] * B[(size_t)k * Nd + j];
  C[(size_t)i * Nd + j] = s;
}

__global__ void copy_param(const float* __restrict__ p, float* __restrict__ o, int D) {
  int i = blockIdx.x * blockDim.x + threadIdx.x;
  if (i < D) o[i] = p[i];
}

// ---------------------------------------------------------------------------
// Tensor Data Mover: DMA a 2D tile (tile_d0 x tile_d1 f32 elements, row stride
// stride0 elements) from global memory into LDS at byte offset lds_off.
// LDS rows are padded +4 DWORDs after every 32 DWORDs (pad_interval=4,
// pad_amount=3) to reproduce the 36-float padded row stride of the LDS tiles.
// ---------------------------------------------------------------------------
__device__ __forceinline__ void tdm_load_2d(unsigned lds_off, const void* gptr,
                                            unsigned tile_d0, unsigned tile_d1,
                                            unsigned tensor_d0, unsigned tensor_d1,
                                            unsigned stride0) {
  unsigned long long ga = (unsigned long long)gptr;
  v4u g0;
  g0[0] = 1u;                                             // count=1 (valid user D#)
  g0[1] = lds_off;                                        // lds_addr
  g0[2] = (unsigned)(ga & 0xFFFFFFFFu);                   // global_addr[31:0]
  g0[3] = (unsigned)((ga >> 32) & 0x01FFFFFFu) | (2u << 30);  // ga[56:32] | type=2
  v8u g1;
  g1[0] = (2u << 16)      // data_size = 2 -> 4-byte elements
        | (1u << 20)      // pad_enable
        | (4u << 22)      // pad_interval: pad after 32 DWORDs
        | (3u << 25);     // pad_amount: +4 DWORDs
  g1[1] = (tensor_d0 & 0xFFFFu) << 16;                          // tensor_dim0[15:0]
  g1[2] = ((tensor_d0 >> 16) & 0xFFFFu) | ((tensor_d1 & 0xFFFFu) << 16);
  g1[3] = ((tensor_d1 >> 16) & 0xFFFFu) | (tile_d0 << 16);      // | tile_dim0
  g1[4] = tile_d1;                                              // tile_dim1 (tile_dim2=0)
  g1[5] = stride0;                                              // tensor_dim0_stride[31:0]
  g1[6] = 0u;                                                   // stride0 hi / stride1 lo
  g1[7] = 0u;
  asm volatile("tensor_load_to_lds %0, %1" :: "s"(g0), "s"(g1) : "memory");
}

__device__ __forceinline__ unsigned lds_byte_off(const void* p) {
  // low 32 bits of a generic pointer into LDS = workgroup-relative LDS offset
  return (unsigned)(unsigned long long)p;
}

// ---------------------------------------------------------------------------
// Main GEMM: Out[B,N] = X[B,K] * W2^T  (W2 is [N,K], N=256, K=512)
// Block: 256 threads (8 waves). Block tile 128x64, K staged 32 via TDM into
// double-buffered LDS. Each wave computes a 32x32 tile = 2x2 WMMA accumulators.
// ---------------------------------------------------------------------------
#define M_BLK 128
#define N_BLK 64
#define K_BLK 32
#define LDP (K_BLK + 4)  // 36 floats = 144 B row stride (TDM pad reproduces this)

__global__ __launch_bounds__(256) void gemm_xw2(const float* __restrict__ X,
                                                const float* __restrict__ W2,
                                                float* __restrict__ Out,
                                                int Bsz, int K, int N) {
  __shared__ float sX[2][M_BLK][LDP];
  __shared__ float sW[2][N_BLK][LDP];

  const int tid   = threadIdx.x;
  const int lane  = tid & 31;
  const int wave  = tid >> 5;   // 0..7
  const int wm    = wave & 3;   // 4 row strips of 32
  const int wn    = wave >> 2;  // 2 col strips of 32
  const int mBase = blockIdx.x * M_BLK;
  const int nBase = blockIdx.y * N_BLK;

  const int lhalf = lane >> 4;  // K-pair select for A/B fragments
  const int l16   = lane & 15;

  const v8f zero = {0.f, 0.f, 0.f, 0.f, 0.f, 0.f, 0.f, 0.f};
  v8f acc[2][2];
  acc[0][0] = zero; acc[0][1] = zero; acc[1][0] = zero; acc[1][1] = zero;

  const unsigned xoff[2] = {lds_byte_off(&sX[0][0][0]), lds_byte_off(&sX[1][0][0])};
  const unsigned woff[2] = {lds_byte_off(&sW[0][0][0]), lds_byte_off(&sW[1][0][0])};
  const float* xTile = X + (size_t)mBase * K;
  const float* wTile = W2 + (size_t)nBase * K;

  const int nst = K / K_BLK;

  // Prologue: stage 0 tiles in flight.
  if (wave == 0) {
    tdm_load_2d(xoff[0], xTile, K_BLK, M_BLK, K, Bsz, K);
    tdm_load_2d(woff[0], wTile, K_BLK, N_BLK, K, N, K);
  }

  for (int s = 0; s < nst; ++s) {
    const int buf = s & 1;
    if (wave == 0) {
      if (s + 1 < nst) {
        const int k1 = (s + 1) * K_BLK;
        tdm_load_2d(xoff[buf ^ 1], xTile + k1, K_BLK, M_BLK, K, Bsz, K);
        tdm_load_2d(woff[buf ^ 1], wTile + k1, K_BLK, N_BLK, K, N, K);
        __builtin_amdgcn_s_wait_tensorcnt((short)2);  // stage s done; s+1 in flight
      } else {
        __builtin_amdgcn_s_wait_tensorcnt((short)0);
      }
    }
    __syncthreads();  // stage-s tiles visible to all waves

#pragma unroll
    for (int kk = 0; kk < K_BLK; kk += 4) {
      const int kf = kk + 2 * lhalf;  // ISA 16x4 A layout: lanes 0-15 K={0,1}, 16-31 K={2,3}
      v2f a[2], b[2];
#pragma unroll
      for (int t = 0; t < 2; ++t) {
        a[t] = *(const v2f*)(&sX[buf][wm * 32 + t * 16 + l16][kf]);
        b[t] = *(const v2f*)(&sW[buf][wn * 32 + t * 16 + l16][kf]);
      }
#pragma unroll
      for (int tm = 0; tm < 2; ++tm)
#pragma unroll
        for (int tn = 0; tn < 2; ++tn)
          acc[tm][tn] = __builtin_amdgcn_wmma_f32_16x16x4_f32(
              false, a[tm], false, b[tn], (short)0, acc[tm][tn], false, false);
    }
    __syncthreads();  // all reads of buf done before stage s+2 DMA overwrites it
  }

  // Epilogue: 16x16 f32 C/D layout — VGPR r: lanes 0-15 -> M=r, lanes 16-31 -> M=r+8
#pragma unroll
  for (int tm = 0; tm < 2; ++tm) {
#pragma unroll
    for (int tn = 0; tn < 2; ++tn) {
      const int col = nBase + wn * 32 + tn * 16 + l16;
#pragma unroll
      for (int r = 0; r < 8; ++r) {
        const int row = mBase + wm * 32 + tm * 16 + r + 8 * lhalf;
        Out[(size_t)row * N + col] = acc[tm][tn][r];
      }
    }
  }
}

// ---------------------------------------------------------------------------
extern "C" void kernel_launch(void* const* d_in, const int* in_sizes, int n_in,
                              void* d_out, int out_size, void* d_ws, size_t ws_size,
                              hipStream_t stream) {
  const float* x     = (const float*)d_in[0];
  const float* Wx    = (const float*)d_in[1];
  const float* Wy    = (const float*)d_in[2];
  const float* param = (const float*)d_in[3];

  const int D   = in_sizes[3];            // 256 (out_dim)
  const int IND = in_sizes[1] / D;        // 512 (in_dim)
  const int Bsz = in_sizes[0] / IND;      // 32768

  float* m0 = (float*)d_ws;               // M ping
  float* m1 = m0 + (size_t)D * D;         // M pong
  float* t0 = m1 + (size_t)D * D;         // T ping
  float* t1 = t0 + (size_t)D * D;         // T pong
  float* W2 = t1 + (size_t)D * D;         // folded weights [D, IND]

  dim3 blk(256);

  // M = I + L, T = L
  init_MT<<<dim3(D / 16, D / 16), blk, 0, stream>>>(Wy, m0, t0, D);

  // 7 doubling steps: T <- T^2 ; M <- M + M*T   ==>  M = sum_{i<256} L^i = (I-L)^-1
  float *M = m0, *Mn = m1, *T = t0, *Tn = t1;
  for (int it = 0; it < 7; ++it) {
    matmul_nn<<<dim3(D / 16, D / 16), blk, 0, stream>>>(T, T, Tn, D, D, D, 0);
    matmul_nn<<<dim3(D / 16, D / 16), blk, 0, stream>>>(M, Tn, Mn, D, D, D, 1);
    float* tmp = T; T = Tn; Tn = tmp;
    tmp = M; M = Mn; Mn = tmp;
  }

  // W2 = M * Wx   [D x IND]
  matmul_nn<<<dim3(IND / 16, D / 16), blk, 0, stream>>>(M, Wx, W2, D, IND, D, 0);

  // out = x * W2^T  — streaming WMMA GEMM, TDM double-buffered LDS staging
  gemm_xw2<<<dim3(Bsz / M_BLK, D / N_BLK), blk, 0, stream>>>(x, W2, (float*)d_out,
                                                             Bsz, IND, D);

  // param passthrough (second tuple element)
  copy_param<<<dim3(1), dim3(D), 0, stream>>>(param, (float*)d_out + (size_t)Bsz * D, D);
}